// MLinear_39599598469144
// MI455X (gfx1250) — compile-verified
//
#include <hip/hip_runtime.h>

typedef __attribute__((ext_vector_type(2))) float v2f;
typedef __attribute__((ext_vector_type(4))) float v4f;
typedef __attribute__((ext_vector_type(8))) float v8f;

#define BM 128
#define BN 128
#define BK 32
#define LDA_S 36   // padded LDS stride for A (floats): gcd(36,64)=4 -> conflict-free row reads
#define LDB_S 136  // padded LDS stride for B (floats): row+2 offset lands 16 banks away

// C[M, outColOff : outColOff+Nblk] = X[:, 0:K] @ W[K, Nblk] + bias, all f32.
// X is [8192 x 4096] row-major, W is [K x ldw] row-major, Out is [8192 x 4096].
__global__ __launch_bounds__(256)
void mlinear_block_gemm_f32(const float* __restrict__ X,
                            const float* __restrict__ W,
                            const float* __restrict__ bias,
                            float* __restrict__ Out,
                            int K, int ldw, int outColOff)
{
    __shared__ float As[BM * LDA_S];
    __shared__ float Bs[BK * LDB_S];

    constexpr int lda = 4096;
    constexpr int ldo = 4096;

    const int tid  = threadIdx.x;      // 0..255 (8 waves of 32)
    const int wave = tid >> 5;
    const int lane = tid & 31;
    const int hl   = lane >> 4;        // half-wave select (0/1)
    const int ll   = lane & 15;

    const int tileM = blockIdx.x * BM;
    const int tileN = blockIdx.y * BN;

    // 2 waves along M x 4 waves along N; each wave: 64 rows x 32 cols (4x2 16x16 tiles)
    const int wm = (wave & 1) * 64;
    const int wn = (wave >> 1) * 32;

    v8f acc[4][2] = {};

    // Global->LDS staging assignments (no bounds checks: all dims are tile multiples)
    const int arow = tid >> 1;          // 128 rows, 2 threads/row, 16 floats each
    const int acol = (tid & 1) * 16;
    const int brow = tid >> 3;          // 32 rows, 8 threads/row, 16 floats each
    const int bcol = (tid & 7) * 16;

    const float* Ag = X + (long)(tileM + arow) * lda + acol;
    const float* Bg = W + (long)brow * ldw + tileN + bcol;

    for (int k0 = 0; k0 < K; k0 += BK) {
        // Load next tiles to registers first (overlaps with LDS drain)
        v4f at[4], bt[4];
        #pragma unroll
        for (int i = 0; i < 4; ++i)
            at[i] = *(const v4f*)(Ag + k0 + 4 * i);
        #pragma unroll
        for (int i = 0; i < 4; ++i)
            bt[i] = *(const v4f*)(Bg + (long)k0 * ldw + 4 * i);

        __syncthreads();   // previous iteration's LDS readers done
        #pragma unroll
        for (int i = 0; i < 4; ++i)
            *(v4f*)(&As[arow * LDA_S + acol + 4 * i]) = at[i];
        #pragma unroll
        for (int i = 0; i < 4; ++i)
            *(v4f*)(&Bs[brow * LDB_S + bcol + 4 * i]) = bt[i];
        __syncthreads();

        #pragma unroll
        for (int kk = 0; kk < BK; kk += 4) {
            // A 16x4 frag: lanes 0-15 hold (K=kk,kk+1), lanes 16-31 hold (K=kk+2,kk+3)
            v2f af[4];
            #pragma unroll
            for (int i = 0; i < 4; ++i)
                af[i] = *(const v2f*)(&As[(wm + 16 * i + ll) * LDA_S + kk + 2 * hl]);

            // B 4x16 frag: mirrored K split, N striped across lanes
            v2f bf[2];
            #pragma unroll
            for (int j = 0; j < 2; ++j) {
                const int c  = wn + 16 * j + ll;
                const int kb = kk + 2 * hl;
                bf[j].x = Bs[kb * LDB_S + c];
                bf[j].y = Bs[(kb + 1) * LDB_S + c];
            }

            #pragma unroll
            for (int i = 0; i < 4; ++i) {
                #pragma unroll
                for (int j = 0; j < 2; ++j) {
                    acc[i][j] = __builtin_amdgcn_wmma_f32_16x16x4_f32(
                        /*neg_a=*/false, af[i],
                        /*neg_b=*/false, bf[j],
                        /*c_mod=*/(short)0, acc[i][j],
                        /*reuse_a=*/false, /*reuse_b=*/false);
                }
            }
        }
    }

    // Epilogue: C/D layout -> row = 8*hl + r, col = ll within each 16x16 tile
    #pragma unroll
    for (int j = 0; j < 2; ++j) {
        const int col = tileN + wn + 16 * j + ll;
        const float bv = bias[outColOff + col];
        #pragma unroll
        for (int i = 0; i < 4; ++i) {
            #pragma unroll
            for (int r = 0; r < 8; ++r) {
                const int row = tileM + wm + 16 * i + 8 * hl + r;
                Out[(long)row * ldo + outColOff + col] = acc[i][j][r] + bv;
            }
        }
    }
}

extern "C" void kernel_launch(void* const* d_in, const int* in_sizes, int n_in,
                              void* d_out, int out_size, void* d_ws, size_t ws_size,
                              hipStream_t stream) {
    const float* x    = (const float*)d_in[0];   // [8192, 4096]
    const float* w0   = (const float*)d_in[1];   // [1024, 1024]
    const float* w1   = (const float*)d_in[2];   // [2048, 1024]
    const float* w2   = (const float*)d_in[3];   // [4096, 2048]
    const float* bias = (const float*)d_in[4];   // [4096]
    float* out = (float*)d_out;                  // [8192, 4096]

    const int M = 8192;
    dim3 block(256);

    // Block-triangular structure: three dense GEMMs into disjoint column ranges.
    mlinear_block_gemm_f32<<<dim3(M / BM, 1024 / BN), block, 0, stream>>>(
        x, w0, bias, out, /*K=*/1024, /*ldw=*/1024, /*outColOff=*/0);
    mlinear_block_gemm_f32<<<dim3(M / BM, 1024 / BN), block, 0, stream>>>(
        x, w1, bias, out, /*K=*/2048, /*ldw=*/1024, /*outColOff=*/1024);
    mlinear_block_gemm_f32<<<dim3(M / BM, 2048 / BN), block, 0, stream>>>(
        x, w2, bias, out, /*K=*/4096, /*ldw=*/2048, /*outColOff=*/2048);
}